// MyViT_65352222376226
// MI455X (gfx1250) — compile-verified
//
#include <hip/hip_runtime.h>
#include <hip/hip_bf16.h>
#include <math.h>

typedef float v2f __attribute__((ext_vector_type(2)));
typedef float v8f __attribute__((ext_vector_type(8)));

#define D_    512
#define SEQ_  511
#define S_    512
#define NB    16
#define HH    8
#define DH_   64
#define MLP_  2048
#define OUTD  51

// ---------------------------------------------------------------------------
// Embedding: tokens = images @ W_map.T + b_map ; prepend cls ; add pos_embed
// ---------------------------------------------------------------------------
__global__ void embed_kernel(const float* __restrict__ images,
                             const float* __restrict__ Wmap,
                             const float* __restrict__ bmap,
                             const float* __restrict__ cls,
                             const float* __restrict__ pos,
                             float* __restrict__ out) {
  int idx = blockIdx.x * blockDim.x + threadIdx.x;
  if (idx >= NB * S_ * D_) return;
  int d = idx & (D_ - 1);
  int s = (idx / D_) & (S_ - 1);
  int n = idx / (D_ * S_);
  float v;
  if (s == 0) {
    v = cls[d];
  } else {
    const float* im = images + ((size_t)n * SEQ_ + (s - 1)) * 2;
    v = im[0] * Wmap[d * 2 + 0] + im[1] * Wmap[d * 2 + 1] + bmap[d];
  }
  out[idx] = v + pos[s * D_ + d];
}

// ---------------------------------------------------------------------------
// LayerNorm over rows of length 512. One block (128 threads) per row.
// ---------------------------------------------------------------------------
__global__ __launch_bounds__(128)
void ln512_kernel(const float* __restrict__ x,
                  const float* __restrict__ g,
                  const float* __restrict__ b,
                  float* __restrict__ y) {
  int row = blockIdx.x;
  int t = threadIdx.x;
  const float* xr = x + (size_t)row * D_;
  float4 v = *(const float4*)(xr + t * 4);
  float s  = v.x + v.y + v.z + v.w;
  float sq = v.x * v.x + v.y * v.y + v.z * v.z + v.w * v.w;
  __shared__ float rs[128], rq[128];
  rs[t] = s; rq[t] = sq;
  __syncthreads();
  for (int o = 64; o > 0; o >>= 1) {
    if (t < o) { rs[t] += rs[t + o]; rq[t] += rq[t + o]; }
    __syncthreads();
  }
  float mean = rs[0] * (1.0f / 512.0f);
  float var  = rq[0] * (1.0f / 512.0f) - mean * mean;
  float inv  = rsqrtf(var + 1e-5f);
  float4 gg = *(const float4*)(g + t * 4);
  float4 bb = *(const float4*)(b + t * 4);
  float4 o4;
  o4.x = (v.x - mean) * inv * gg.x + bb.x;
  o4.y = (v.y - mean) * inv * gg.y + bb.y;
  o4.z = (v.z - mean) * inv * gg.z + bb.z;
  o4.w = (v.w - mean) * inv * gg.w + bb.w;
  *(float4*)(y + (size_t)row * D_ + t * 4) = o4;
}

// ---------------------------------------------------------------------------
// Cooperative staging of a 128x16 A tile and (64*NSUB)x16 W tile into LDS.
// Row stride 20 floats: float4-store aligned, bank-conflict-free frag loads.
// ---------------------------------------------------------------------------
template <int NSUB>
__device__ __forceinline__
void stage_tiles(const float* __restrict__ A, int lda,
                 const float* __restrict__ W, int ldw,
                 int m0, int n0, int k0,
                 float* __restrict__ Al, float* __restrict__ Wl, int t) {
  #pragma unroll
  for (int i = 0; i < 2; ++i) {                 // 512 float4 slots for A
    int id = t + i * 256;
    int row = id >> 2, cq = (id & 3) * 4;
    *(float4*)&Al[row * 20 + cq] =
        *(const float4*)(A + (size_t)(m0 + row) * lda + k0 + cq);
  }
  #pragma unroll
  for (int i = 0; i < NSUB; ++i) {              // 256*NSUB float4 slots for W
    int id = t + i * 256;
    int row = id >> 2, cq = (id & 3) * 4;
    *(float4*)&Wl[row * 20 + cq] =
        *(const float4*)(W + (size_t)(n0 + row) * ldw + k0 + cq);
  }
}

// ---------------------------------------------------------------------------
// Generic fp32 WMMA GEMM:  C = act(alpha * A @ W^T + bias) (+ res)
//   A: M x K (lda), W: N x K (ldw), C: M x N (ldc). Batched via blockIdx.z.
//   Block tile 128(M) x 64*NSUB(N); 8 waves (4x2); wave tile 32 x 32*NSUB.
//   Double-buffered LDS (one barrier per 16-wide K chunk) + global prefetch.
//   Requires M%128==0, N%(64*NSUB)==0, K%16==0 (holds at all call sites).
//   op: 0 = none, 1 = exact GELU.
// ---------------------------------------------------------------------------
template <int NSUB>
__global__ __launch_bounds__(256)
void gemm_wmma_f32(const float* __restrict__ A, int lda, long long strideA,
                   const float* __restrict__ W, int ldw, long long strideW,
                   const float* __restrict__ bias, long long strideB,
                   const float* __restrict__ res,
                   float* __restrict__ C, int ldc, long long strideC,
                   int K, float alpha, int op) {
  A += (size_t)blockIdx.z * strideA;
  W += (size_t)blockIdx.z * strideW;
  C += (size_t)blockIdx.z * strideC;
  if (bias) bias += (size_t)blockIdx.z * strideB;
  if (res)  res  += (size_t)blockIdx.z * strideC;

  const int NT = 32 * NSUB;                 // wave tile width in N
  const int m0 = blockIdx.y * 128;
  const int n0 = blockIdx.x * (64 * NSUB);

  __shared__ float Al[2][128 * 20];
  __shared__ float Wl[2][64 * NSUB * 20];

  const int t = threadIdx.x;
  const int lane = t & 31;
  const int wave = t >> 5;
  const int wm = wave >> 1;                 // 0..3  (M, 32 rows each)
  const int wn = wave & 1;                  // 0..1  (N, NT cols each)
  const int mf = lane & 15;
  const int kq = (lane >> 4) * 2;           // lanes 0-15 -> K0/K1, 16-31 -> K2/K3

  v8f acc[2][2 * NSUB] = {};

  const int nchunk = K >> 4;
  stage_tiles<NSUB>(A, lda, W, ldw, m0, n0, 0, Al[0], Wl[0], t);

  for (int c = 0; c < nchunk; ++c) {
    __syncthreads();                        // staged buffer (c&1) ready
    const int cur = c & 1;
    if (c + 1 < nchunk)
      stage_tiles<NSUB>(A, lda, W, ldw, m0, n0, (c + 1) * 16,
                        Al[cur ^ 1], Wl[cur ^ 1], t);
    if (c + 2 < nchunk) {                   // global_prefetch for chunk c+2
      __builtin_prefetch(A + (size_t)(m0 + (t & 127)) * lda + (c + 2) * 16, 0, 1);
      __builtin_prefetch(W + (size_t)(n0 + (t & (64 * NSUB - 1))) * ldw + (c + 2) * 16, 0, 1);
    }
    const float* al = Al[cur];
    const float* wl = Wl[cur];
    #pragma unroll
    for (int kk = 0; kk < 4; ++kk) {
      int ko = kk * 4 + kq;
      v2f a0 = *(const v2f*)&al[(wm * 32 + mf) * 20 + ko];
      v2f a1 = *(const v2f*)&al[(wm * 32 + 16 + mf) * 20 + ko];
      #pragma unroll
      for (int ni = 0; ni < 2 * NSUB; ++ni) {
        v2f b = *(const v2f*)&wl[(wn * NT + ni * 16 + mf) * 20 + ko];
        acc[0][ni] = __builtin_amdgcn_wmma_f32_16x16x4_f32(
            false, a0, false, b, (short)0, acc[0][ni], false, false);
        acc[1][ni] = __builtin_amdgcn_wmma_f32_16x16x4_f32(
            false, a1, false, b, (short)0, acc[1][ni], false, false);
      }
    }
  }

  // Epilogue. C/D layout: lane col = lane&15, VGPR r -> row r + 8*(lane>>4).
  const int mhi = (lane >> 4) * 8;
  #pragma unroll
  for (int mi = 0; mi < 2; ++mi) {
    #pragma unroll
    for (int ni = 0; ni < 2 * NSUB; ++ni) {
      int nn = n0 + wn * NT + ni * 16 + mf;
      float bv = bias ? bias[nn] : 0.0f;
      #pragma unroll
      for (int r = 0; r < 8; ++r) {
        int mm = m0 + wm * 32 + mi * 16 + mhi + r;
        float v = alpha * acc[mi][ni][r] + bv;
        if (op == 1) v = 0.5f * v * (1.0f + erff(v * 0.7071067811865476f));
        size_t off = (size_t)mm * ldc + nn;
        if (res) v += res[off];
        C[off] = v;
      }
    }
  }
}

// ---------------------------------------------------------------------------
// QKV projections for the (single) attended sequence.
//   grid.z = h*3 + which (which: 0=Q, 1=K, 2=V-transposed)
//   A = xn0[:, h*64 : h*64+64] (512x64, lda=512); W = Wsel[h] (64x64)
//   Block tile 64(M) x 64(N); wave tile 16 x 32. Direct global loads (tiny K).
// ---------------------------------------------------------------------------
__global__ __launch_bounds__(256)
void qkv_kernel(const float* __restrict__ xn0,
                const float* __restrict__ Wq, const float* __restrict__ bq,
                const float* __restrict__ Wk, const float* __restrict__ bk,
                const float* __restrict__ Wv, const float* __restrict__ bv,
                float* __restrict__ Q, float* __restrict__ Kb,
                float* __restrict__ VT) {
  int z = blockIdx.z;
  int h = z / 3, which = z % 3;
  const float* Wsel = (which == 0) ? Wq : (which == 1) ? Wk : Wv;
  const float* bsel = (which == 0) ? bq : (which == 1) ? bk : bv;
  Wsel += (size_t)h * DH_ * DH_;
  bsel += (size_t)h * DH_;

  const int t = threadIdx.x, lane = t & 31, wave = t >> 5;
  const int wm = wave >> 1, wn = wave & 1;        // 4 x 2 waves
  const int mf = lane & 15, kq = (lane >> 4) * 2;
  const int m0 = blockIdx.x * 64;

  const float* arow = xn0 + (size_t)(m0 + wm * 16 + mf) * D_ + h * DH_;
  v8f acc[2] = {};
  #pragma unroll
  for (int kk = 0; kk < 16; ++kk) {
    int ko = kk * 4 + kq;
    v2f a  = *(const v2f*)(arow + ko);
    v2f b0 = *(const v2f*)(Wsel + (size_t)(wn * 32 + mf) * DH_ + ko);
    v2f b1 = *(const v2f*)(Wsel + (size_t)(wn * 32 + 16 + mf) * DH_ + ko);
    acc[0] = __builtin_amdgcn_wmma_f32_16x16x4_f32(false, a, false, b0, (short)0, acc[0], false, false);
    acc[1] = __builtin_amdgcn_wmma_f32_16x16x4_f32(false, a, false, b1, (short)0, acc[1], false, false);
  }
  const int mhi = (lane >> 4) * 8;
  #pragma unroll
  for (int ni = 0; ni < 2; ++ni) {
    int n = wn * 32 + ni * 16 + mf;
    float bvv = bsel[n];
    #pragma unroll
    for (int r = 0; r < 8; ++r) {
      int mm = m0 + wm * 16 + mhi + r;
      float v = acc[ni][r] + bvv;
      if (which == 0)      Q [(size_t)h * S_ * DH_ + (size_t)mm * DH_ + n] = v;
      else if (which == 1) Kb[(size_t)h * S_ * DH_ + (size_t)mm * DH_ + n] = v;
      else                 VT[(size_t)h * DH_ * S_ + (size_t)n * S_ + mm] = v;  // transposed
    }
  }
}

// ---------------------------------------------------------------------------
// In-place softmax over rows of length 512. One block (256 threads) per row.
// ---------------------------------------------------------------------------
__global__ __launch_bounds__(256)
void softmax512_kernel(float* __restrict__ sc) {
  int row = blockIdx.x, t = threadIdx.x;
  float* p = sc + (size_t)row * S_;
  float a = p[t], b = p[t + 256];
  __shared__ float red[256];
  red[t] = fmaxf(a, b);
  __syncthreads();
  for (int o = 128; o > 0; o >>= 1) {
    if (t < o) red[t] = fmaxf(red[t], red[t + o]);
    __syncthreads();
  }
  float mx = red[0];
  __syncthreads();
  float ea = expf(a - mx), eb = expf(b - mx);
  red[t] = ea + eb;
  __syncthreads();
  for (int o = 128; o > 0; o >>= 1) {
    if (t < o) red[t] += red[t + o];
    __syncthreads();
  }
  float inv = 1.0f / red[0];
  p[t] = ea * inv;
  p[t + 256] = eb * inv;
}

// ---------------------------------------------------------------------------
// out[n, s, d] += msa[s, d]  broadcast over batch
// ---------------------------------------------------------------------------
__global__ void addbcast_kernel(float* __restrict__ out,
                                const float* __restrict__ msa) {
  int i = blockIdx.x * blockDim.x + threadIdx.x;
  if (i >= S_ * D_) return;
  float m = msa[i];
  #pragma unroll
  for (int n = 0; n < NB; ++n) out[(size_t)n * S_ * D_ + i] += m;
}

// ---------------------------------------------------------------------------
// Fused head MLP per row: 512 ->16 relu ->32 relu ->64 relu ->51
// ---------------------------------------------------------------------------
__global__ __launch_bounds__(128)
void head_mlp_kernel(const float* __restrict__ X,
                     const float* __restrict__ Wm1, const float* __restrict__ bm1,
                     const float* __restrict__ Wm2, const float* __restrict__ bm2,
                     const float* __restrict__ Wm3, const float* __restrict__ bm3,
                     const float* __restrict__ Wm4, const float* __restrict__ bm4,
                     float* __restrict__ Y) {
  __shared__ float x[D_];
  __shared__ float h1[16], h2[32], h3[64];
  int row = blockIdx.x, t = threadIdx.x;
  *(float4*)&x[t * 4] = *(const float4*)(X + (size_t)row * D_ + t * 4);
  __syncthreads();
  {  // h1: 16 outputs, 8 lanes each (within-wave groups of 8)
    int o = t >> 3, g = t & 7;
    const float* w = Wm1 + o * D_;
    float s = 0.0f;
    for (int k = g; k < D_; k += 8) s += w[k] * x[k];
    s += __shfl_down(s, 4, 8);
    s += __shfl_down(s, 2, 8);
    s += __shfl_down(s, 1, 8);
    if (g == 0) h1[o] = fmaxf(s + bm1[o], 0.0f);
  }
  __syncthreads();
  if (t < 32) {
    const float* w = Wm2 + t * 16;
    float s = bm2[t];
    #pragma unroll
    for (int k = 0; k < 16; ++k) s += w[k] * h1[k];
    h2[t] = fmaxf(s, 0.0f);
  }
  __syncthreads();
  if (t < 64) {
    const float* w = Wm3 + t * 32;
    float s = bm3[t];
    #pragma unroll
    for (int k = 0; k < 32; ++k) s += w[k] * h2[k];
    h3[t] = fmaxf(s, 0.0f);
  }
  __syncthreads();
  if (t < OUTD) {
    const float* w = Wm4 + t * 64;
    float s = bm4[t];
    #pragma unroll
    for (int k = 0; k < 64; ++k) s += w[k] * h3[k];
    Y[(size_t)row * OUTD + t] = s;
  }
}

// ---------------------------------------------------------------------------
extern "C" void kernel_launch(void* const* d_in, const int* in_sizes, int n_in,
                              void* d_out, int out_size, void* d_ws, size_t ws_size,
                              hipStream_t stream) {
  const float* images = (const float*)d_in[0];
  const float* W_map  = (const float*)d_in[1];
  const float* b_map  = (const float*)d_in[2];
  const float* cls    = (const float*)d_in[3];
  const float* pos    = (const float*)d_in[4];
  const float* ln1_g  = (const float*)d_in[5];
  const float* ln1_b  = (const float*)d_in[6];
  const float* Wq     = (const float*)d_in[7];
  const float* bq     = (const float*)d_in[8];
  const float* Wk     = (const float*)d_in[9];
  const float* bk     = (const float*)d_in[10];
  const float* Wv     = (const float*)d_in[11];
  const float* bv     = (const float*)d_in[12];
  const float* ln2_g  = (const float*)d_in[13];
  const float* ln2_b  = (const float*)d_in[14];
  const float* W1     = (const float*)d_in[15];
  const float* b1     = (const float*)d_in[16];
  const float* W2     = (const float*)d_in[17];
  const float* b2     = (const float*)d_in[18];
  const float* Wm1    = (const float*)d_in[19];
  const float* bm1    = (const float*)d_in[20];
  const float* Wm2    = (const float*)d_in[21];
  const float* bm2    = (const float*)d_in[22];
  const float* Wm3    = (const float*)d_in[23];
  const float* bm3    = (const float*)d_in[24];
  const float* Wm4    = (const float*)d_in[25];
  const float* bm4    = (const float*)d_in[26];

  float* ws = (float*)d_ws;
  float* OUTB = ws;                          // 16 * 512 * 512        = 4,194,304 f
  float* YN   = ws + 4194304;                // 4,194,304 f
  float* BIG  = ws + 8388608;                // 16,777,216 f (h1 / attn scratch)
  float* XN0  = BIG;                         //   512*512   = 262,144
  float* Qb   = BIG + 262144;                //   8*512*64  = 262,144
  float* Kb   = BIG + 524288;                //   262,144
  float* VT   = BIG + 786432;                //   262,144 (per-head 64x512)
  float* MSA  = BIG + 1048576;               //   262,144
  float* SC   = BIG + 1310720;               //   8*512*512 = 2,097,152
  float* H1   = BIG;                         // 8192 x 2048 (after attention done)

  embed_kernel<<<(NB * S_ * D_ + 255) / 256, 256, 0, stream>>>(
      images, W_map, b_map, cls, pos, OUTB);

  for (int l = 0; l < 2; ++l) {
    // LN1 — only batch 0 is ever used by the (faithfully buggy) attention.
    ln512_kernel<<<S_, 128, 0, stream>>>(OUTB, ln1_g + l * D_, ln1_b + l * D_, XN0);

    qkv_kernel<<<dim3(S_ / 64, 1, 3 * HH), 256, 0, stream>>>(
        XN0,
        Wq + (size_t)l * HH * DH_ * DH_, bq + (size_t)l * HH * DH_,
        Wk + (size_t)l * HH * DH_ * DH_, bk + (size_t)l * HH * DH_,
        Wv + (size_t)l * HH * DH_ * DH_, bv + (size_t)l * HH * DH_,
        Qb, Kb, VT);

    // scores[h] = (Q_h @ K_h^T) / 8   (M=512, N=512, K=64, batch=8)
    gemm_wmma_f32<2><<<dim3(S_ / 128, S_ / 128, HH), 256, 0, stream>>>(
        Qb, DH_, (long long)S_ * DH_,
        Kb, DH_, (long long)S_ * DH_,
        nullptr, 0, nullptr,
        SC, S_, (long long)S_ * S_,
        DH_, 0.125f, 0);

    softmax512_kernel<<<HH * S_, 256, 0, stream>>>(SC);

    // msa[:, h*64:(h+1)*64] = attn_h @ V_h   (A @ VT^T, M=512, N=64, K=512)
    gemm_wmma_f32<1><<<dim3(DH_ / 64, S_ / 128, HH), 256, 0, stream>>>(
        SC, S_, (long long)S_ * S_,
        VT, S_, (long long)DH_ * S_,
        nullptr, 0, nullptr,
        MSA, D_, (long long)DH_,
        S_, 1.0f, 0);

    addbcast_kernel<<<(S_ * D_ + 255) / 256, 256, 0, stream>>>(OUTB, MSA);

    ln512_kernel<<<NB * S_, 128, 0, stream>>>(OUTB, ln2_g + l * D_, ln2_b + l * D_, YN);

    // h1 = gelu(yn @ W1^T + b1)   (M=8192, N=2048, K=512)
    gemm_wmma_f32<2><<<dim3(MLP_ / 128, NB * S_ / 128, 1), 256, 0, stream>>>(
        YN, D_, 0,
        W1 + (size_t)l * MLP_ * D_, D_, 0,
        b1 + (size_t)l * MLP_, 0, nullptr,
        H1, MLP_, 0,
        D_, 1.0f, 1);

    // out = out + h1 @ W2^T + b2   (M=8192, N=512, K=2048, in-place residual)
    gemm_wmma_f32<2><<<dim3(D_ / 128, NB * S_ / 128, 1), 256, 0, stream>>>(
        H1, MLP_, 0,
        W2 + (size_t)l * D_ * MLP_, MLP_, 0,
        b2 + (size_t)l * D_, 0, OUTB,
        OUTB, D_, 0,
        MLP_, 1.0f, 0);
  }

  head_mlp_kernel<<<NB * S_, 128, 0, stream>>>(
      OUTB, Wm1, bm1, Wm2, bm2, Wm3, bm3, Wm4, bm4, (float*)d_out);
}